// FCOS_42090679501461
// MI455X (gfx1250) — compile-verified
//
#include <hip/hip_runtime.h>
#include <math.h>
#include <stdint.h>

#define TPB 256
#define MAXM 128

// FCOS GT matching + delta/centerness + box decode.
// Grid: x = location blocks (per batch), y = batch. One thread per location.
// GT boxes for the block's batch are broadcast into LDS via CDNA5 async
// global->LDS loads (ASYNCcnt path), repacked to SoA float4, then reused
// 256x by the 128-box matching loop (ds_load_b128 broadcast, 2 DS ops/box).
__global__ __launch_bounds__(TPB) void fcos_match_kernel(
    const float* __restrict__ locs3, const float* __restrict__ locs4,
    const float* __restrict__ locs5, const float* __restrict__ gt,
    const float* __restrict__ p3, const float* __restrict__ p4,
    const float* __restrict__ p5, float* __restrict__ out,
    int N3, int N4, int N5, int NT, int M)
{
    __shared__ __align__(16) float  s_raw[MAXM * 5];
    __shared__ __align__(16) float4 s_box[MAXM];
    __shared__ float s_qv[MAXM];

    const int tid = threadIdx.x;
    const int b   = blockIdx.y;                    // batch index (no division)
    const int n   = blockIdx.x * TPB + tid;        // concatenated location idx

    // ---- CDNA5 async global->LDS broadcast of this batch's GT table ----
    {
        const float* gtb = gt + (size_t)b * (size_t)M * 5;
        const int nchunks = (M * 5) >> 2;          // 16B chunks (M%4==0)
        const unsigned lds_base = (unsigned)(uintptr_t)(&s_raw[0]);
        for (int c = tid; c < nchunks; c += TPB) {
            const unsigned lds_addr = lds_base + (unsigned)c * 16u;
            const unsigned long long ga =
                (unsigned long long)(uintptr_t)(gtb + (size_t)c * 4);
            asm volatile("global_load_async_to_lds_b128 %0, %1, off"
                         :: "v"(lds_addr), "v"(ga) : "memory");
        }
        asm volatile("s_wait_asynccnt 0" ::: "memory");
    }
    __syncthreads();

    // SoA repack + per-box match quality (1e8 - area).
    for (int m = tid; m < M; m += TPB) {
        const float x0 = s_raw[m * 5 + 0];
        const float y0 = s_raw[m * 5 + 1];
        const float x1 = s_raw[m * 5 + 2];
        const float y1 = s_raw[m * 5 + 3];
        s_box[m] = make_float4(x0, y0, x1, y1);
        s_qv[m]  = 1e8f - (x1 - x0) * (y1 - y0);
    }
    __syncthreads();

    if (n >= NT) return;

    // Level selection (n is the concatenated per-batch location index).
    const float* locs; const float* pred;
    float stride, lo, hi; int nl, NL;
    if (n < N3) {
        locs = locs3; pred = p3; stride = 8.f;  lo = 0.f;   hi = 64.f;
        nl = n;            NL = N3;
    } else if (n < N3 + N4) {
        locs = locs4; pred = p4; stride = 16.f; lo = 64.f;  hi = 128.f;
        nl = n - N3;       NL = N4;
    } else {
        locs = locs5; pred = p5; stride = 32.f; lo = 128.f; hi = INFINITY;
        nl = n - N3 - N4;  NL = N5;
    }

    const float2 xy = *reinterpret_cast<const float2*>(locs + 2 * (size_t)nl);
    const float x = xy.x;
    const float y = xy.y;

    // ---- match against M GT boxes (first-max argmax semantics) ----
    float bestq = 0.f; int besti = 0;
#pragma unroll 4
    for (int m = 0; m < M; ++m) {
        const float4 bx = s_box[m];                 // ds_load_b128 broadcast
        const float l = x - bx.x;
        const float t = y - bx.y;
        const float r = bx.z - x;
        const float d = bx.w - y;
        const float pmin = fminf(fminf(l, t), fminf(r, d));
        const float pmax = fmaxf(fmaxf(l, t), fmaxf(r, d));
        const bool ok = (pmin > 0.f) & (pmax > lo) & (pmax < hi);
        const float q = ok ? s_qv[m] : 0.f;
        if (q > bestq) { bestq = q; besti = m; }    // strict > keeps first max
    }

    // ---- deltas + centerness for the matched box ----
    float d0, d1, d2, d3, ctr;
    if (bestq < 1e-5f) {
        d0 = d1 = d2 = d3 = -1.f;
        ctr = -1.f;
    } else {
        const float4 bb = s_box[besti];
        const float inv = 1.f / stride;             // exact (stride = 2^k)
        d0 = (x - bb.x) * inv;
        d1 = (y - bb.y) * inv;
        d2 = (bb.z - x) * inv;
        d3 = (bb.w - y) * inv;
        const float lrmin = fminf(d0, d2), lrmax = fmaxf(d0, d2);
        const float tbmin = fminf(d1, d3), tbmax = fmaxf(d1, d3);
        const float ratio = fminf(lrmin, tbmin) / (fmaxf(lrmax, tbmax) + 1e-6f);
        ctr = sqrtf(fmaxf(ratio, 0.f));
    }

    // ---- decode predicted boxes (dc*stride is exact: stride = 2^k) ----
    const float4 pv =
        *reinterpret_cast<const float4*>(pred + ((size_t)b * NL + nl) * 4);
    const float dc0 = fmaxf(pv.x, 0.f), dc1 = fmaxf(pv.y, 0.f);
    const float dc2 = fmaxf(pv.z, 0.f), dc3 = fmaxf(pv.w, 0.f);
    const float ex1 = fmaxf(x - dc0 * stride, 0.f);
    const float ey1 = fmaxf(y - dc1 * stride, 0.f);
    const float ex2 = fmaxf(x + dc2 * stride, 0.f);
    const float ey2 = fmaxf(y + dc3 * stride, 0.f);

    float* o = out + ((size_t)b * NT + n) * 9;
    o[0] = d0;  o[1] = d1;  o[2] = d2;  o[3] = d3;
    o[4] = ctr;
    o[5] = ex1; o[6] = ey1; o[7] = ex2; o[8] = ey2;
}

extern "C" void kernel_launch(void* const* d_in, const int* in_sizes, int n_in,
                              void* d_out, int out_size, void* d_ws, size_t ws_size,
                              hipStream_t stream) {
    (void)n_in; (void)out_size; (void)d_ws; (void)ws_size;
    const float* locs3 = (const float*)d_in[0];
    const float* locs4 = (const float*)d_in[1];
    const float* locs5 = (const float*)d_in[2];
    const float* gt    = (const float*)d_in[3];
    const float* p3    = (const float*)d_in[4];
    const float* p4    = (const float*)d_in[5];
    const float* p5    = (const float*)d_in[6];

    const int N3 = in_sizes[0] / 2;
    const int N4 = in_sizes[1] / 2;
    const int N5 = in_sizes[2] / 2;
    const int NT = N3 + N4 + N5;
    const int B  = in_sizes[4] / (N3 * 4);        // pred_p3 = B*N3*4
    int       M  = in_sizes[3] / (B * 5);         // gt = B*M*5
    if (M > MAXM) M = MAXM;

    dim3 grid((unsigned)((NT + TPB - 1) / TPB), (unsigned)B, 1);
    fcos_match_kernel<<<grid, TPB, 0, stream>>>(
        locs3, locs4, locs5, gt, p3, p4, p5, (float*)d_out,
        N3, N4, N5, NT, M);
}